// SignedDistance_57698590655052
// MI455X (gfx1250) — compile-verified
//
#include <hip/hip_runtime.h>
#include <stdint.h>

// SignedDistance: memory-bound gather/stream kernel for MI455X (gfx1250).
// CDNA5 async global->LDS b128 copies (ASYNCcnt) double-buffer the streamed
// per-point inputs; LDS is the ONLY source for compute (tail handled by
// clamping async addresses), so ds_load must be emitted. Triangle/normal
// gathers stay L2-resident; outputs use non-temporal stores.

#define TILE      256              // points per tile == block size (8 waves, wave32)
#define NCHUNK_P  192              // TILE*12 bytes / 16 = b128 chunks for points/bcs
#define NCHUNK_I  64               // TILE*4  bytes / 16 = b128 chunks for faces

// Flat pointer to LDS: aperture maps addr[31:0] -> group-relative LDS offset.
__device__ __forceinline__ uint32_t lds_off(const void* p) {
    return (uint32_t)(uintptr_t)p;
}

__device__ __forceinline__ void async_b128(uint32_t lds, uint64_t gaddr) {
    asm volatile("global_load_async_to_lds_b128 %0, %1, off"
                 :: "v"(lds), "v"(gaddr)
                 : "memory");
}

__global__ __launch_bounds__(TILE)
void SignedDistance_57698590655052_kernel(
    const float* __restrict__ tris,   // F*9  (L2-resident, gathered)
    const float* __restrict__ nrm,    // F*3  (L2-resident, gathered)
    const float* __restrict__ pts,    // Q*3  (streamed via async->LDS)
    const int*   __restrict__ faces,  // Q    (streamed via async->LDS)
    const float* __restrict__ bcs,    // Q*3  (streamed via async->LDS)
    float* __restrict__ out,
    int Q, int numTiles)
{
    __shared__ __attribute__((aligned(16))) float s_pts[2][TILE * 3];
    __shared__ __attribute__((aligned(16))) float s_bcs[2][TILE * 3];
    __shared__ __attribute__((aligned(16))) int   s_idx[2][TILE];

    float* out_sd = out;                          // [0, Q)
    float* out_rn = out + (size_t)Q;              // [Q, 4Q)
    float* out_cp = out + (size_t)4 * Q;          // [4Q, 7Q)
    int*   out_cf = (int*)(out + (size_t)7 * Q);  // [7Q, 8Q) int bits
    float* out_bc = out + (size_t)8 * Q;          // [8Q, 11Q)

    const int tid    = threadIdx.x;
    const int stride = gridDim.x;
    int tile = blockIdx.x;

    // Last in-bounds 16B-aligned chunk offset per array (tail chunks clamp here;
    // duplicate loads of the same chunk to stale LDS slots are benign).
    const uint64_t maxP = (((uint64_t)Q * 12u) - 16u) & ~15ull;
    const uint64_t maxI = (((uint64_t)Q * 4u)  - 16u) & ~15ull;

    // Exactly 3 async b128 loads per thread per tile -> per-wave ASYNCcnt += 3.
    auto issue = [&](int t, int b) {
        const uint64_t start = (uint64_t)t * TILE;
        const uint32_t cp = (tid < NCHUNK_P) ? (uint32_t)tid : (uint32_t)(tid - NCHUNK_P);
        uint64_t offP = start * 12u + (uint64_t)cp * 16u;
        if (offP > maxP) offP = maxP;
        async_b128(lds_off(&s_pts[b][0]) + cp * 16u, (uint64_t)(uintptr_t)pts + offP);
        async_b128(lds_off(&s_bcs[b][0]) + cp * 16u, (uint64_t)(uintptr_t)bcs + offP);

        const uint32_t ci = (uint32_t)(tid & (NCHUNK_I - 1));
        uint64_t offI = start * 4u + (uint64_t)ci * 16u;
        if (offI > maxI) offI = maxI;
        async_b128(lds_off(&s_idx[b][0]) + ci * 16u, (uint64_t)(uintptr_t)faces + offI);
    };

    // Prologue: prefetch first tile into buffer 0.
    if (tile < numTiles) issue(tile, 0);

    int cur = 0;
    for (; tile < numTiles; tile += stride, cur ^= 1) {
        const int  next    = tile + stride;
        const bool hasNext = (next < numTiles);

        if (hasNext) issue(next, cur ^ 1);   // prefetch next while computing cur

        if (hasNext) {
            // 3 (cur) + 3 (next) in flight; async loads retire in order per
            // wave, so <=3 means cur's copies have landed in LDS.
            asm volatile("s_wait_asynccnt 3" ::: "memory");
        } else {
            asm volatile("s_wait_asynccnt 0" ::: "memory");
        }
        __syncthreads();   // all waves' copies for buf[cur] visible block-wide

        const long long q = (long long)tile * TILE + tid;
        if (q < (long long)Q) {
            // LDS is the only source: ds_load path
            const float px = s_pts[cur][tid * 3 + 0];
            const float py = s_pts[cur][tid * 3 + 1];
            const float pz = s_pts[cur][tid * 3 + 2];
            const float b0 = s_bcs[cur][tid * 3 + 0];
            const float b1 = s_bcs[cur][tid * 3 + 1];
            const float b2 = s_bcs[cur][tid * 3 + 2];
            const int   f  = s_idx[cur][tid];

            // clip barycentrics to [0,1]
            const float w0 = fminf(fmaxf(b0, 0.0f), 1.0f);
            const float w1 = fminf(fmaxf(b1, 0.0f), 1.0f);
            const float w2 = fminf(fmaxf(b2, 0.0f), 1.0f);

            // gather triangle (36 B) and normal (12 B) — L2-resident working set
            const float* tv = tris + (size_t)f * 9;
            const float t00 = tv[0], t01 = tv[1], t02 = tv[2];
            const float t10 = tv[3], t11 = tv[4], t12 = tv[5];
            const float t20 = tv[6], t21 = tv[7], t22 = tv[8];

            // closest point: sum_v tri[v][c] * w[v]
            const float cpx = t00 * w0 + t10 * w1 + t20 * w2;
            const float cpy = t01 * w0 + t11 * w1 + t21 * w2;
            const float cpz = t02 * w0 + t12 * w1 + t22 * w2;

            const float rx = cpx - px;
            const float ry = cpy - py;
            const float rz = cpz - pz;

            const float d2   = rx * rx + ry * ry + rz * rz;
            const float dist = sqrtf(d2);
            const float nv   = (dist == 0.0f) ? 1.0f : dist;
            const float rnx  = rx / nv;
            const float rny  = ry / nv;
            const float rnz  = rz / nv;

            const float* fnv = nrm + (size_t)f * 3;
            const float dp = rnx * fnv[0] + rny * fnv[1] + rnz * fnv[2];
            const float sg = (dp > 0.0f) ? -1.0f : 1.0f;
            const float sd = sg * dist;

            // write-only outputs: non-temporal so triangle data stays hot in L2
            __builtin_nontemporal_store(sd,  &out_sd[q]);
            __builtin_nontemporal_store(rnx, &out_rn[q * 3 + 0]);
            __builtin_nontemporal_store(rny, &out_rn[q * 3 + 1]);
            __builtin_nontemporal_store(rnz, &out_rn[q * 3 + 2]);
            __builtin_nontemporal_store(cpx, &out_cp[q * 3 + 0]);
            __builtin_nontemporal_store(cpy, &out_cp[q * 3 + 1]);
            __builtin_nontemporal_store(cpz, &out_cp[q * 3 + 2]);
            __builtin_nontemporal_store(f,   &out_cf[q]);
            __builtin_nontemporal_store(w0,  &out_bc[q * 3 + 0]);
            __builtin_nontemporal_store(w1,  &out_bc[q * 3 + 1]);
            __builtin_nontemporal_store(w2,  &out_bc[q * 3 + 2]);
        }
        __syncthreads();   // buf[cur] fully consumed before next iter overwrites it
    }
}

extern "C" void kernel_launch(void* const* d_in, const int* in_sizes, int n_in,
                              void* d_out, int out_size, void* d_ws, size_t ws_size,
                              hipStream_t stream) {
    const float* tris  = (const float*)d_in[0];   // (1,F,3,3) f32
    const float* fnrm  = (const float*)d_in[1];   // (1,F,3)   f32
    const float* pts   = (const float*)d_in[2];   // (1,Q,3)   f32
    const int*   faces = (const int*)  d_in[3];   // (1,Q)     i32
    const float* bcs   = (const float*)d_in[4];   // (1,Q,3)   f32

    const int Q        = in_sizes[3];             // B*Q, B == 1
    const int numTiles = (Q + TILE - 1) / TILE;
    int grid = numTiles < 2048 ? numTiles : 2048; // persistent blocks -> real
                                                  // async double-buffer pipelining
    SignedDistance_57698590655052_kernel<<<grid, TILE, 0, stream>>>(
        tris, fnrm, pts, faces, bcs, (float*)d_out, Q, numTiles);
}